// PPOTrainer_35708358099469
// MI455X (gfx1250) — compile-verified
//
#include <hip/hip_runtime.h>

typedef float v2f __attribute__((ext_vector_type(2)));
typedef float v8f __attribute__((ext_vector_type(8)));

#define TT    262144
#define SS    512
#define ACT   64
#define GAMMAF 0.99f
#define LAMF   0.98f
#define CLIPF  0.2f
#define ENTCF  0.01f
#define LDSW  72          // padded LDS row stride (floats): 2*72 mod 64 = 16 -> conflict-free halves
#define NCH   4096        // GAE chunks
#define CLEN  64          // elements per chunk (NCH*CLEN == TT)

// ---------------------------------------------------------------------------
// Row post-processing: log-softmax over 64 cols spread as 4 N-tiles x 16 lanes
// ---------------------------------------------------------------------------
__device__ __forceinline__ void process_tile(
    const v8f (&acc)[4], int tbase, int half, int lm,
    float b0, float b1, float b2, float b3,
    const int* __restrict__ actions, const float* __restrict__ log_probs,
    float* __restrict__ wn, float& ent_local)
{
#pragma unroll
  for (int j = 0; j < 8; ++j) {
    float l0 = acc[0][j] + b0;
    float l1 = acc[1][j] + b1;
    float l2 = acc[2][j] + b2;
    float l3 = acc[3][j] + b3;
    // row max across 16 lanes of this half
    float m = fmaxf(fmaxf(l0, l1), fmaxf(l2, l3));
#pragma unroll
    for (int off = 1; off < 16; off <<= 1) m = fmaxf(m, __shfl_xor(m, off));
    float e0 = expf(l0 - m), e1 = expf(l1 - m), e2 = expf(l2 - m), e3 = expf(l3 - m);
    float s    = e0 + e1 + e2 + e3;
    float enum_ = e0 * l0 + e1 * l1 + e2 * l2 + e3 * l3;
#pragma unroll
    for (int off = 1; off < 16; off <<= 1) {
      s     += __shfl_xor(s, off);
      enum_ += __shfl_xor(enum_, off);
    }
    float logZ = m + logf(s);

    int r = tbase + j + 8 * half;
    int a = actions[r];
    int an = a >> 4;
    float sel = (lm == (a & 15))
                    ? (an == 0 ? l0 : (an == 1 ? l1 : (an == 2 ? l2 : l3)))
                    : 0.f;
#pragma unroll
    for (int off = 1; off < 16; off <<= 1) sel += __shfl_xor(sel, off);

    if (lm == j) {
      wn[r] = (sel - logZ) - log_probs[r];          // new_log_prob - old_log_prob
      ent_local += (logZ - enum_ / s);              // entropy of this row
    }
  }
}

// ---------------------------------------------------------------------------
// Kernel 1: fused actor GEMM (WMMA f32 16x16x4) + critic GEMVs + softmax stats
// Each wave: 32 rows (2 M-tiles) x 64 cols (4 N-tiles), K = 512.
// ---------------------------------------------------------------------------
__global__ __launch_bounds__(256) void ppo_main(
    const float* __restrict__ states, const float* __restrict__ next_states,
    const float* __restrict__ rewards, const int* __restrict__ dones,
    const int* __restrict__ actions, const float* __restrict__ log_probs,
    const float* __restrict__ actor_w, const float* __restrict__ actor_b,
    const float* __restrict__ critic_w, const float* __restrict__ critic_b,
    float* __restrict__ wd, float* __restrict__ wc, float* __restrict__ wn,
    float* __restrict__ accum)
{
  __shared__ float s_aw[128 * LDSW];   // one 128-row K-chunk of actor_w, padded
  __shared__ float s_cw[SS];           // critic_w

  const int tid  = threadIdx.x;
  const int wave = tid >> 5;
  const int lane = tid & 31;
  const int half = lane >> 4;
  const int lm   = lane & 15;
  const int t0   = blockIdx.x * 256 + wave * 32;
  const int t1   = t0 + 16;

  if (tid < 128) *(float4*)&s_cw[tid * 4] = ((const float4*)critic_w)[tid];

  v8f acc0[4], acc1[4];
#pragma unroll
  for (int n = 0; n < 4; ++n) {
    acc0[n] = (v8f){0.f, 0.f, 0.f, 0.f, 0.f, 0.f, 0.f, 0.f};
    acc1[n] = (v8f){0.f, 0.f, 0.f, 0.f, 0.f, 0.f, 0.f, 0.f};
  }
  float cacc0 = 0.f, cacc1 = 0.f, nacc0 = 0.f, nacc1 = 0.f;

  for (int kc = 0; kc < 4; ++kc) {
    __syncthreads();
    // stage 128x64 chunk of actor_w into padded LDS (coalesced float4)
#pragma unroll
    for (int it = 0; it < 8; ++it) {
      int idx = tid + it * 256;
      int row = idx >> 4, q = idx & 15;
      float4 v = ((const float4*)(actor_w + (size_t)(kc * 128 + row) * ACT))[q];
      *(float4*)&s_aw[row * LDSW + q * 4] = v;
    }
    __syncthreads();

    for (int kk = 0; kk < 32; ++kk) {
      const int k0 = kc * 128 + kk * 4;   // global k base
      const int kr = kk * 4;              // k within LDS chunk
      const int kcol = k0 + 2 * half;
      const int krl  = kr + 2 * half;

      // B fragments (4x16 of actor_w) from LDS, shared by both M-tiles
      v2f bfr[4];
#pragma unroll
      for (int n = 0; n < 4; ++n) {
        bfr[n].x = s_aw[(krl + 0) * LDSW + 16 * n + lm];
        bfr[n].y = s_aw[(krl + 1) * LDSW + 16 * n + lm];
      }
      // A fragments (16x4 of states) + matching next_states pieces
      v2f a0 = *(const v2f*)(states      + (size_t)(t0 + lm) * SS + kcol);
      v2f a1 = *(const v2f*)(states      + (size_t)(t1 + lm) * SS + kcol);
      v2f q0 = *(const v2f*)(next_states + (size_t)(t0 + lm) * SS + kcol);
      v2f q1 = *(const v2f*)(next_states + (size_t)(t1 + lm) * SS + kcol);

      float cw0 = s_cw[kcol], cw1 = s_cw[kcol + 1];
      cacc0 += a0.x * cw0 + a0.y * cw1;  cacc1 += a1.x * cw0 + a1.y * cw1;
      nacc0 += q0.x * cw0 + q0.y * cw1;  nacc1 += q1.x * cw0 + q1.y * cw1;

#pragma unroll
      for (int n = 0; n < 4; ++n) {
        acc0[n] = __builtin_amdgcn_wmma_f32_16x16x4_f32(
            false, a0, false, bfr[n], (short)0, acc0[n], false, false);
        acc1[n] = __builtin_amdgcn_wmma_f32_16x16x4_f32(
            false, a1, false, bfr[n], (short)0, acc1[n], false, false);
      }
    }
  }

  // critic values: combine lane halves (each holds half of K)
  const float cb = critic_b[0];
  float v0  = cacc0 + __shfl_xor(cacc0, 16) + cb;
  float v1  = cacc1 + __shfl_xor(cacc1, 16) + cb;
  float nv0 = nacc0 + __shfl_xor(nacc0, 16) + cb;
  float nv1 = nacc1 + __shfl_xor(nacc1, 16) + cb;

  if (half == 0) {
    int r = t0 + lm;
    float nd = 1.f - (float)dones[r];
    wd[r] = rewards[r] + GAMMAF * nv0 * nd - v0;
    wc[r] = (GAMMAF * LAMF) * nd;
    r = t1 + lm;
    nd = 1.f - (float)dones[r];
    wd[r] = rewards[r] + GAMMAF * nv1 * nd - v1;
    wc[r] = (GAMMAF * LAMF) * nd;
  }

  float b0 = actor_b[lm], b1 = actor_b[16 + lm], b2 = actor_b[32 + lm], b3 = actor_b[48 + lm];
  float ent_local = 0.f;
  process_tile(acc0, t0, half, lm, b0, b1, b2, b3, actions, log_probs, wn, ent_local);
  process_tile(acc1, t1, half, lm, b0, b1, b2, b3, actions, log_probs, wn, ent_local);

#pragma unroll
  for (int off = 1; off < 32; off <<= 1) ent_local += __shfl_xor(ent_local, off);
  if (lane == 0) atomicAdd(&accum[0], ent_local);
}

// ---------------------------------------------------------------------------
// Kernel 2a: per-chunk affine composition  adv_start = A + B * adv_end
// ---------------------------------------------------------------------------
__global__ void gae_chunk(const float* __restrict__ wd, const float* __restrict__ wc,
                          float* __restrict__ chA, float* __restrict__ chB)
{
  int g = blockIdx.x * blockDim.x + threadIdx.x;    // 0..NCH-1
  int base = g * CLEN;
  float Ac = 0.f, Bc = 1.f;
  for (int i = CLEN - 1; i >= 0; --i) {
    float di = wd[base + i], ci = wc[base + i];
    Ac = di + ci * Ac;
    Bc = ci * Bc;
  }
  chA[g] = Ac;
  chB[g] = Bc;
}

// ---------------------------------------------------------------------------
// Kernel 2b: two-level suffix scan of chunk functions -> per-chunk carry-in
// ---------------------------------------------------------------------------
__global__ __launch_bounds__(256) void gae_scan(const float* __restrict__ chA,
                                                const float* __restrict__ chB,
                                                float* __restrict__ advIn)
{
  __shared__ float sA[NCH], sB[NCH];
  __shared__ float supA[256], supB[256], carryIn[256];
  const int t = threadIdx.x;
#pragma unroll
  for (int m = 0; m < 16; ++m) {
    int idx = t + m * 256;
    sA[idx] = chA[idx];
    sB[idx] = chB[idx];
  }
  __syncthreads();
  // compose 16 chunk functions into one superchunk function (right to left)
  float Ac = 0.f, Bc = 1.f;
  for (int m = 15; m >= 0; --m) {
    int g = t * 16 + m;
    Ac = sA[g] + sB[g] * Ac;
    Bc = sB[g] * Bc;
  }
  supA[t] = Ac;
  supB[t] = Bc;
  __syncthreads();
  if (t == 0) {
    float carry = 0.f;
    for (int u = 255; u >= 0; --u) {
      carryIn[u] = carry;
      carry = supA[u] + supB[u] * carry;
    }
  }
  __syncthreads();
  float carry = carryIn[t];
  for (int m = 15; m >= 0; --m) {
    int g = t * 16 + m;
    advIn[g] = carry;                  // adv at first element of chunk g+1
    carry = sA[g] + sB[g] * carry;
  }
}

// ---------------------------------------------------------------------------
// Kernel 2c: replay chunks with carries; accumulate PPO loss reductions
// ---------------------------------------------------------------------------
__global__ void gae_final(const float* __restrict__ wd, const float* __restrict__ wc,
                          const float* __restrict__ wn, const float* __restrict__ advIn,
                          float* __restrict__ accum)
{
  int g = blockIdx.x * blockDim.x + threadIdx.x;    // 0..NCH-1
  int base = g * CLEN;
  float carry = advIn[g];
  float sum_min = 0.f, sum_adv2 = 0.f;
  for (int i = CLEN - 1; i >= 0; --i) {
    float adv = wd[base + i] + wc[base + i] * carry;
    carry = adv;
    float ratio = expf(wn[base + i]);
    float s1 = ratio * adv;
    float rc = fminf(fmaxf(ratio, 1.f - CLIPF), 1.f + CLIPF);
    float s2 = rc * adv;
    sum_min  += fminf(s1, s2);
    sum_adv2 += adv * adv;             // (values - returns)^2 == adv^2
  }
#pragma unroll
  for (int off = 1; off < 32; off <<= 1) {
    sum_min  += __shfl_xor(sum_min, off);
    sum_adv2 += __shfl_xor(sum_adv2, off);
  }
  if ((threadIdx.x & 31) == 0) {
    atomicAdd(&accum[1], sum_min);
    atomicAdd(&accum[2], sum_adv2);
  }
}

// ---------------------------------------------------------------------------
// Kernel 3: finalize the two scalar losses
// ---------------------------------------------------------------------------
__global__ void ppo_finalize(const float* __restrict__ accum, float* __restrict__ out)
{
  const float invT = 1.f / (float)TT;
  out[0] = -(accum[1] * invT) - ENTCF * (accum[0] * invT);   // actor_loss
  out[1] = accum[2] * invT;                                  // value_loss
}

// ---------------------------------------------------------------------------
extern "C" void kernel_launch(void* const* d_in, const int* in_sizes, int n_in,
                              void* d_out, int out_size, void* d_ws, size_t ws_size,
                              hipStream_t stream)
{
  (void)in_sizes; (void)n_in; (void)out_size; (void)ws_size;
  const float* states      = (const float*)d_in[0];
  const float* next_states = (const float*)d_in[1];
  const float* rewards     = (const float*)d_in[2];
  const int*   dones       = (const int*)d_in[3];
  const int*   actions     = (const int*)d_in[4];
  const float* log_probs   = (const float*)d_in[5];
  const float* actor_w     = (const float*)d_in[6];
  const float* actor_b     = (const float*)d_in[7];
  const float* critic_w    = (const float*)d_in[8];
  const float* critic_b    = (const float*)d_in[9];

  float* ws    = (float*)d_ws;
  float* wd    = ws;                 // deltas            [TT]
  float* wc    = ws + TT;            // gamma*lam*nd      [TT]
  float* wn    = ws + 2 * TT;        // nlp - old_logp    [TT]
  float* chA   = ws + 3 * TT;        // chunk A           [NCH]
  float* chB   = chA + NCH;          // chunk B           [NCH]
  float* advIn = chB + NCH;          // chunk carry-in    [NCH]
  float* accum = advIn + NCH;        // [ent, sum_min, sum_adv2]

  hipMemsetAsync(accum, 0, 3 * sizeof(float), stream);

  ppo_main<<<TT / 256, 256, 0, stream>>>(states, next_states, rewards, dones,
                                         actions, log_probs, actor_w, actor_b,
                                         critic_w, critic_b, wd, wc, wn, accum);
  gae_chunk<<<NCH / 256, 256, 0, stream>>>(wd, wc, chA, chB);
  gae_scan<<<1, 256, 0, stream>>>(chA, chB, advIn);
  gae_final<<<NCH / 256, 256, 0, stream>>>(wd, wc, wn, advIn, accum);
  ppo_finalize<<<1, 1, 0, stream>>>(accum, (float*)d_out);
}